// Net_74388833567279
// MI455X (gfx1250) — compile-verified
//
#include <hip/hip_runtime.h>
#include <hip/hip_bf16.h>
#include <math.h>

typedef float v2f __attribute__((ext_vector_type(2)));
typedef float v8f __attribute__((ext_vector_type(8)));

__device__ __forceinline__ void atomicAddF(float* p, float v) {
    // non-returning fp32 atomic -> global_atomic_add_f32 (STOREcnt tracked)
    __hip_atomic_fetch_add(p, v, __ATOMIC_RELAXED, __HIP_MEMORY_SCOPE_AGENT);
}

// ---------------- degree / normalization ----------------
__global__ void k_degree(const long long* __restrict__ dst, float* __restrict__ deg, int E) {
    int e = blockIdx.x * blockDim.x + threadIdx.x;
    if (e < E) atomicAddF(&deg[(int)dst[e]], 1.0f);
}

__global__ void k_dinv(const float* __restrict__ deg, float* __restrict__ dinv, int n) {
    int i = blockIdx.x * blockDim.x + threadIdx.x;
    if (i < n) dinv[i] = rsqrtf(deg[i] + 1.0f);   // self-loop included
}

// ---------------- layer-1 zero padding (K=5 -> K=8) ----------------
__global__ void k_pad_x(const float* __restrict__ x, float* __restrict__ x8, int n) {
    int idx = blockIdx.x * blockDim.x + threadIdx.x;      // over n*8
    if (idx >= n * 8) return;
    const int i = idx >> 3, c = idx & 7;
    x8[idx] = (c < 5) ? x[i * 5 + c] : 0.0f;
}

__global__ void k_pad_w1(const float* __restrict__ W1, float* __restrict__ W1p) {
    int idx = threadIdx.x;                                // 8*32 = 256
    W1p[idx] = (idx < 5 * 32) ? W1[idx] : 0.0f;
}

// ---------------- fp32 WMMA GEMM:  C[n x M] = A[n x K] @ W[K x M] ----------------
// One wave (32 threads) per 16x16 output tile; K and M compile-time so every stride is
// an immediate: the k-loop is a branch-free unrolled stream of clause-grouped
// global_load_b64/b32 + v_wmma_f32_16x16x4_f32, and the store tail is 8 immediate-offset
// global_store_b32 with no exec manipulation on the (always-taken) full-tile path.
// A 16x4 layout: lanes 0-15 hold K={k,k+1}, lanes 16-31 hold K={k+2,k+3}, row = lane&15.
// C 16x16 layout: VGPR r -> row (half*8 + r), col = lane&15.
template <int K, int M>
__global__ void k_gemm_wmma(const float* __restrict__ A, const float* __restrict__ W,
                            float* __restrict__ C, int n) {
    const int lane = threadIdx.x & 31;
    const int half = lane >> 4;
    const int l    = lane & 15;
    const int row0 = blockIdx.x * 16;
    const int col0 = blockIdx.y * 16;

    int rowA = row0 + l;
    if (rowA >= n) rowA = n - 1;          // uniform clamp; EXEC stays all-ones
    const float* __restrict__ Arow = A + (size_t)rowA * K + half * 2;   // 8B aligned
    const float* __restrict__ Wp   = W + (size_t)(half * 2) * M + col0 + l;

    v8f acc = {0.f, 0.f, 0.f, 0.f, 0.f, 0.f, 0.f, 0.f};

#pragma unroll
    for (int k = 0; k < K; k += 4) {
        const float2 av = *(const float2*)(Arow + k);     // global_load_b64
        v2f a, b;
        a.x = av.x;
        a.y = av.y;
        b.x = Wp[k * M];                                  // W[k + 2*half    ][col]  (imm offset)
        b.y = Wp[k * M + M];                              // W[k + 2*half + 1][col]  (imm offset)
        acc = __builtin_amdgcn_wmma_f32_16x16x4_f32(
                  /*neg_a=*/false, a, /*neg_b=*/false, b,
                  /*c_mod=*/(short)0, acc, /*reuse_a=*/false, /*reuse_b=*/false);
    }

    const int rbase = row0 + half * 8;
    float* __restrict__ p = C + (size_t)rbase * M + col0 + l;
    if (row0 + 15 < n) {
        // full tile (always the case for N % 16 == 0): unguarded immediate-offset stores
#pragma unroll
        for (int r = 0; r < 8; ++r) p[r * M] = acc[r];
    } else {
#pragma unroll
        for (int r = 0; r < 8; ++r) {
            if (rbase + r < n) p[r * M] = acc[r];
        }
    }
}

// ---------------- edge scatter: agg[dst] += h[src] * dinv[src]*dinv[dst] ----------------
// thread = (edge, 4-channel chunk); adjacent threads cover adjacent channels -> coalesced atomics.
__global__ void k_scatter(const long long* __restrict__ src, const long long* __restrict__ dst,
                          const float* __restrict__ h, const float* __restrict__ dinv,
                          float* __restrict__ agg, long long total, int M, int lchunks) {
    long long tid = (long long)blockIdx.x * blockDim.x + threadIdx.x;
    if (tid >= total) return;
    const int chunks = M >> 2;
    const int e = (int)(tid >> lchunks);
    const int c = ((int)tid & (chunks - 1)) << 2;
    const int s = (int)src[e];
    const int d = (int)dst[e];
    const float norm = dinv[s] * dinv[d];
    const float4 v = *(const float4*)(h + (size_t)s * M + c);
    float* p = agg + (size_t)d * M + c;
    atomicAddF(p + 0, v.x * norm);
    atomicAddF(p + 1, v.y * norm);
    atomicAddF(p + 2, v.z * norm);
    atomicAddF(p + 3, v.w * norm);
}

// ---------------- self-loop + bias + relu (in place on agg) ----------------
__global__ void k_self_bias_relu(float* __restrict__ agg, const float* __restrict__ h,
                                 const float* __restrict__ dinv, const float* __restrict__ bias,
                                 long long total, int M, int lM) {
    long long idx = (long long)blockIdx.x * blockDim.x + threadIdx.x;
    if (idx >= total) return;
    const int i = (int)(idx >> lM);
    const int c = (int)idx & (M - 1);
    const float di = dinv[i];
    const float v = agg[idx] + h[idx] * (di * di) + bias[c];
    agg[idx] = fmaxf(v, 0.0f);
}

// ---------------- global mean pool ----------------
__global__ void k_pool_sum(const float* __restrict__ h, const long long* __restrict__ batch,
                           float* __restrict__ sums, long long total) {
    long long idx = (long long)blockIdx.x * blockDim.x + threadIdx.x;
    if (idx >= total) return;
    const int i = (int)(idx >> 8);      // M == 256 on the last layer
    const int c = (int)idx & 255;
    atomicAddF(&sums[(size_t)batch[i] * 256 + c], h[idx]);
}

__global__ void k_pool_cnt(const long long* __restrict__ batch, float* __restrict__ cnt, int n) {
    int i = blockIdx.x * blockDim.x + threadIdx.x;
    if (i < n) atomicAddF(&cnt[(int)batch[i]], 1.0f);
}

// ---------------- FC head: pooled -> fc1(relu) -> fc2 -> log_softmax ----------------
__global__ void k_head(const float* __restrict__ sums, const float* __restrict__ cnt,
                       const float* __restrict__ fc1_w, const float* __restrict__ fc1_b,
                       const float* __restrict__ fc2_w, const float* __restrict__ fc2_b,
                       float* __restrict__ out) {
    const int b   = blockIdx.x;
    const int tid = threadIdx.x;
    __shared__ float pooled[256];
    __shared__ float z1[100];
    __shared__ float z2[10];

    const float inv = 1.0f / fmaxf(cnt[b], 1.0f);
    for (int i = tid; i < 256; i += blockDim.x) pooled[i] = sums[(size_t)b * 256 + i] * inv;
    __syncthreads();

    if (tid < 100) {
        float s = fc1_b[tid];
        for (int k = 0; k < 256; ++k) s = fmaf(pooled[k], fc1_w[(size_t)k * 100 + tid], s);
        z1[tid] = fmaxf(s, 0.0f);
    }
    __syncthreads();

    if (tid < 10) {
        float s = fc2_b[tid];
        for (int j = 0; j < 100; ++j) s = fmaf(z1[j], fc2_w[(size_t)j * 10 + tid], s);
        z2[tid] = s;
    }
    __syncthreads();

    if (tid == 0) {
        float m = z2[0];
        for (int t = 1; t < 10; ++t) m = fmaxf(m, z2[t]);
        float se = 0.0f;
        for (int t = 0; t < 10; ++t) se += expf(z2[t] - m);
        const float lse = m + logf(se);
        for (int t = 0; t < 10; ++t) out[(size_t)b * 10 + t] = z2[t] - lse;
    }
}

extern "C" void kernel_launch(void* const* d_in, const int* in_sizes, int n_in,
                              void* d_out, int out_size, void* d_ws, size_t ws_size,
                              hipStream_t stream) {
    (void)n_in; (void)ws_size;
    const float*     x     = (const float*)d_in[0];
    const long long* ei    = (const long long*)d_in[1];
    const long long* batch = (const long long*)d_in[2];
    const float* Ws[4] = { (const float*)d_in[3], (const float*)d_in[5],
                           (const float*)d_in[7], (const float*)d_in[9] };
    const float* bs[4] = { (const float*)d_in[4], (const float*)d_in[6],
                           (const float*)d_in[8], (const float*)d_in[10] };
    const float* fc1_w = (const float*)d_in[11];
    const float* fc1_b = (const float*)d_in[12];
    const float* fc2_w = (const float*)d_in[13];
    const float* fc2_b = (const float*)d_in[14];

    const int N = in_sizes[0] / 5;
    const int E = in_sizes[1] / 2;
    const int B = out_size / 10;
    const long long* src = ei;
    const long long* dst = ei + E;

    // ---- workspace carve-out (256B aligned) ----
    char*  ws  = (char*)d_ws;
    size_t off = 0;
    auto take = [&](size_t bytes) -> float* {
        float* p = (float*)(ws + off);
        off = (off + bytes + 255) & ~(size_t)255;
        return p;
    };
    float* deg  = take((size_t)N * 4);
    float* dinv = take((size_t)N * 4);
    float* x8   = take((size_t)N * 8 * 4);
    float* w1p  = take((size_t)8 * 32 * 4);
    float* hbuf = take((size_t)N * 256 * 4);
    float* agg  = take((size_t)N * 256 * 4);
    float* sums = take((size_t)B * 256 * 4);
    float* cnt  = take((size_t)B * 4);

    // ---- degrees / symmetric normalization ----
    hipMemsetAsync(deg, 0, (size_t)N * 4, stream);
    k_degree<<<(E + 255) / 256, 256, 0, stream>>>(dst, deg, E);
    k_dinv<<<(N + 255) / 256, 256, 0, stream>>>(deg, dinv, N);

    // ---- pad layer-1 operands to K=8 ----
    k_pad_x<<<(N * 8 + 255) / 256, 256, 0, stream>>>(x, x8, N);
    k_pad_w1<<<1, 256, 0, stream>>>(Ws[0], w1p);

    // ---- 4 GCN layers ----
    const int dims[5] = {5, 32, 64, 128, 256};
    const float* cur = x8;
    for (int l = 0; l < 4; ++l) {
        const int M = dims[l + 1];

        dim3 grid((N + 15) / 16, M / 16);
        switch (l) {
            case 0: k_gemm_wmma<8,   32 ><<<grid, 32, 0, stream>>>(cur, w1p,   hbuf, N); break;
            case 1: k_gemm_wmma<32,  64 ><<<grid, 32, 0, stream>>>(cur, Ws[1], hbuf, N); break;
            case 2: k_gemm_wmma<64,  128><<<grid, 32, 0, stream>>>(cur, Ws[2], hbuf, N); break;
            case 3: k_gemm_wmma<128, 256><<<grid, 32, 0, stream>>>(cur, Ws[3], hbuf, N); break;
        }

        hipMemsetAsync(agg, 0, (size_t)N * M * 4, stream);

        const int lchunks = __builtin_ctz(M >> 2);
        const long long totE = (long long)E * (M >> 2);
        k_scatter<<<(unsigned)((totE + 255) / 256), 256, 0, stream>>>(
            src, dst, hbuf, dinv, agg, totE, M, lchunks);

        const int lM = __builtin_ctz(M);
        const long long totN = (long long)N * M;
        k_self_bias_relu<<<(unsigned)((totN + 255) / 256), 256, 0, stream>>>(
            agg, hbuf, dinv, bs[l], totN, M, lM);

        cur = agg;   // layer output lives in agg; gemm of next layer reads it before memset
    }

    // ---- global mean pool ----
    hipMemsetAsync(sums, 0, (size_t)B * 256 * 4, stream);
    hipMemsetAsync(cnt, 0, (size_t)B * 4, stream);
    const long long totP = (long long)N * 256;
    k_pool_sum<<<(unsigned)((totP + 255) / 256), 256, 0, stream>>>(cur, batch, sums, totP);
    k_pool_cnt<<<(N + 255) / 256, 256, 0, stream>>>(batch, cnt, N);

    // ---- FC head + log_softmax ----
    k_head<<<B, 128, 0, stream>>>(sums, cnt, fc1_w, fc1_b, fc2_w, fc2_b, (float*)d_out);
}